// GCN_27032524161760
// MI455X (gfx1250) — compile-verified
//
#include <hip/hip_runtime.h>
#include <hip/hip_bf16.h>

// ---------------------------------------------------------------------------
// CDNA5 (gfx1250) GCN forward: two GCNConv layers + mean pool.
// GEMMs use V_WMMA_F32_16X16X4_F32 (full fp32 precision on the matrix pipe).
// B tile is staged in LDS pre-interleaved in WMMA operand order so every
// operand fetch is one aligned ds_load_b64 (no register-assembly movs).
// ---------------------------------------------------------------------------

typedef float v2f __attribute__((ext_vector_type(2)));
typedef float v8f __attribute__((ext_vector_type(8)));

#define TM 64
#define TN 128
#define TK 32
#define LDA_PAD 36        // floats; 144B rows -> 16B aligned, conflict-free A fetch
#define PAIR_STRIDE 264   // floats per B half-row (128 pairs = 256 floats + 8 pad)

// ---------------------------------------------------------------------------
// WMMA GEMM: C[M,N] = A[M,K] * B[K,N], row-major fp32. K % 32 == 0, N % 128 == 0.
// Block: 256 threads (8 waves). Block tile 64x128, wave tile 16x64 (4 accs).
// ---------------------------------------------------------------------------
__global__ __launch_bounds__(256) void gcn_gemm_wmma_f32(
    const float* __restrict__ A, const float* __restrict__ B,
    float* __restrict__ C, int M, int K, int N)
{
    __shared__ float sA[TM * LDA_PAD];
    __shared__ float sB[(TK / 2) * PAIR_STRIDE];   // 16 half-rows of interleaved pairs

    const int tid   = threadIdx.x;
    const int lane  = tid & 31;
    const int wave  = tid >> 5;
    const int laneM = lane & 15;   // M-row within 16x16 tile (also N-col for B/C)
    const int kHalf = lane >> 4;   // 0: K={0,1}/rows r ; 1: K={2,3}/rows r+8
    const int waveM = wave & 3;    // 4 waves along M
    const int waveN = wave >> 2;   // 2 waves along N

    const int rowBlock = blockIdx.y * TM;
    const int colBlock = blockIdx.x * TN;

    v8f acc0 = {}, acc1 = {}, acc2 = {}, acc3 = {};

    for (int k0 = 0; k0 < K; k0 += TK) {
        // --- stage A tile: 64 x 32 (2 float4 per thread), zero-pad rows >= M
        #pragma unroll
        for (int i = 0; i < 2; ++i) {
            int v = tid + i * 256;          // 0..511 float4 slots
            int r = v >> 3;                 // 8 float4 per row
            int c = (v & 7) << 2;
            int grow = rowBlock + r;
            float4 val = make_float4(0.f, 0.f, 0.f, 0.f);
            if (grow < M) val = *(const float4*)(A + (size_t)grow * K + k0 + c);
            *(float4*)(sA + r * LDA_PAD + c) = val;
        }
        // --- stage B tile 32x128 pair-interleaved: half-row bk/2 holds
        //     {B[bk][c], B[bk+1][c]} pairs, contiguous in c.
        #pragma unroll
        for (int i = 0; i < 2; ++i) {
            int v  = tid + i * 256;         // 0..511 tasks (16 row-pairs x 32 col-groups)
            int pr = v >> 5;                // row-pair index 0..15  (bk = 2*pr)
            int cg = (v & 31) << 2;         // col 0..124 step 4
            const float* gp = B + (size_t)(k0 + (pr << 1)) * N + colBlock + cg;
            float4 r0 = *(const float4*)(gp);
            float4 r1 = *(const float4*)(gp + N);
            float* sp = sB + pr * PAIR_STRIDE + (cg << 1);
            *(float4*)(sp)     = make_float4(r0.x, r1.x, r0.y, r1.y);
            *(float4*)(sp + 4) = make_float4(r0.z, r1.z, r0.w, r1.w);
        }
        __syncthreads();

        #pragma unroll
        for (int kk = 0; kk < TK; kk += 4) {
            // A operand: lane<16 -> {K=kk,kk+1}, lane>=16 -> {K=kk+2,kk+3}
            v2f a = *(const v2f*)(sA + (waveM * 16 + laneM) * LDA_PAD + kk + 2 * kHalf);
            // B operand: half-row (kk + 2*kHalf)/2, contiguous pair per column
            const float* bp = sB + ((kk >> 1) + kHalf) * PAIR_STRIDE
                            + ((waveN * 64 + laneM) << 1);
            v2f b0 = *(const v2f*)(bp);
            v2f b1 = *(const v2f*)(bp + 32);
            v2f b2 = *(const v2f*)(bp + 64);
            v2f b3 = *(const v2f*)(bp + 96);
            acc0 = __builtin_amdgcn_wmma_f32_16x16x4_f32(false, a, false, b0, (short)0, acc0, false, false);
            acc1 = __builtin_amdgcn_wmma_f32_16x16x4_f32(false, a, false, b1, (short)0, acc1, false, false);
            acc2 = __builtin_amdgcn_wmma_f32_16x16x4_f32(false, a, false, b2, (short)0, acc2, false, false);
            acc3 = __builtin_amdgcn_wmma_f32_16x16x4_f32(false, a, false, b3, (short)0, acc3, false, false);
        }
        __syncthreads();
    }

    // --- store D: VGPR r -> rows (base + r) for lanes<16, (base + r + 8) for lanes>=16
    const int col     = colBlock + waveN * 64 + laneM;
    const int rowBase = rowBlock + waveM * 16 + 8 * kHalf;
    #pragma unroll
    for (int r = 0; r < 8; ++r) {
        int grow = rowBase + r;
        if (grow < M) {
            float* cp = C + (size_t)grow * N + col;
            cp[0]  = acc0[r];
            cp[16] = acc1[r];
            cp[32] = acc2[r];
            cp[48] = acc3[r];
        }
    }
}

// ---------------------------------------------------------------------------
// Degree / normalization helpers
// ---------------------------------------------------------------------------
__global__ void gcn_fill(float* __restrict__ p, int n, float v) {
    int i = blockIdx.x * blockDim.x + threadIdx.x;
    if (i < n) p[i] = v;
}

__global__ void gcn_edge_deg(const int* __restrict__ dst, const float* __restrict__ ew,
                             float* __restrict__ deg, int E) {
    int e = blockIdx.x * blockDim.x + threadIdx.x;
    if (e < E) atomicAdd(deg + dst[e], ew[e]);
}

__global__ void gcn_rsqrt_inplace(float* __restrict__ p, int n) {
    int i = blockIdx.x * blockDim.x + threadIdx.x;
    if (i < n) {
        float d = p[i];
        p[i] = (d > 0.f) ? rsqrtf(d) : 0.f;
    }
}

// out[n][:] = dinv[n]^2 * xw[n][:]   (self-loop term; norm = dinv*1*dinv)
__global__ void gcn_self_init(const float* __restrict__ xw, const float* __restrict__ dinv,
                              float* __restrict__ out, int F) {
    int n = blockIdx.x;
    float c = dinv[n]; c *= c;
    const float4* xp = (const float4*)(xw + (size_t)n * F);
    float4* op = (float4*)(out + (size_t)n * F);
    float4 v = xp[threadIdx.x];
    op[threadIdx.x] = make_float4(c * v.x, c * v.y, c * v.z, c * v.w);
}

// One block per edge: out[dst] += (dinv[src]*ew*dinv[dst]) * xw[src]
__global__ void gcn_edge_scatter(const int* __restrict__ src, const int* __restrict__ dst,
                                 const float* __restrict__ ew, const float* __restrict__ dinv,
                                 const float* __restrict__ xw, float* __restrict__ out, int F) {
    int e = blockIdx.x;
    int s = src[e];
    int d = dst[e];
    float c = dinv[s] * ew[e] * dinv[d];
    const float4* xp = (const float4*)(xw + (size_t)s * F);
    float* op = out + (size_t)d * F;
    float4 v = xp[threadIdx.x];
    int f = threadIdx.x << 2;
    atomicAdd(op + f + 0, c * v.x);
    atomicAdd(op + f + 1, c * v.y);
    atomicAdd(op + f + 2, c * v.z);
    atomicAdd(op + f + 3, c * v.w);
}

__global__ void gcn_bias_relu(float* __restrict__ h, const float* __restrict__ b, int F) {
    int n = blockIdx.x;
    float4* hp = (float4*)(h + (size_t)n * F);
    const float4* bp = (const float4*)b;
    float4 v = hp[threadIdx.x];
    float4 bb = bp[threadIdx.x];
    v.x = fmaxf(v.x + bb.x, 0.f);
    v.y = fmaxf(v.y + bb.y, 0.f);
    v.z = fmaxf(v.z + bb.z, 0.f);
    v.w = fmaxf(v.w + bb.w, 0.f);
    hp[threadIdx.x] = v;
}

// ---------------------------------------------------------------------------
// Mean pool
// ---------------------------------------------------------------------------
__global__ void gcn_pool_accum(const float* __restrict__ h, const int* __restrict__ batch,
                               float* __restrict__ pool, float* __restrict__ cnt, int F) {
    int n = blockIdx.x;
    int g = batch[n];
    if (threadIdx.x == 0) atomicAdd(cnt + g, 1.0f);
    const float4* hp = (const float4*)(h + (size_t)n * F);
    float* pp = pool + (size_t)g * F;
    float4 v = hp[threadIdx.x];
    int f = threadIdx.x << 2;
    atomicAdd(pp + f + 0, v.x);
    atomicAdd(pp + f + 1, v.y);
    atomicAdd(pp + f + 2, v.z);
    atomicAdd(pp + f + 3, v.w);
}

__global__ void gcn_pool_div(const float* __restrict__ pool, const float* __restrict__ cnt,
                             float* __restrict__ out, int F) {
    int g = blockIdx.x;
    float inv = 1.0f / fmaxf(cnt[g], 1.0f);
    const float4* pp = (const float4*)(pool + (size_t)g * F);
    float4* op = (float4*)(out + (size_t)g * F);
    float4 v = pp[threadIdx.x];
    op[threadIdx.x] = make_float4(v.x * inv, v.y * inv, v.z * inv, v.w * inv);
}

// ---------------------------------------------------------------------------
// Host-side launch
// Inputs: 0 node_features[N,Fin] f32, 1 edge_index[2,E] i32, 2 edge_weight[E] f32,
//         3 batch[N] i32, 4 W1[Fin,Fout] f32, 5 b1[Fout], 6 W2[Fout,Fout], 7 b2[Fout]
// Output: [G, Fout] f32
// ---------------------------------------------------------------------------
extern "C" void kernel_launch(void* const* d_in, const int* in_sizes, int n_in,
                              void* d_out, int out_size, void* d_ws, size_t ws_size,
                              hipStream_t stream) {
    const float* x     = (const float*)d_in[0];
    const int*   ei    = (const int*)d_in[1];
    const float* ew    = (const float*)d_in[2];
    const int*   batch = (const int*)d_in[3];
    const float* W1    = (const float*)d_in[4];
    const float* b1    = (const float*)d_in[5];
    const float* W2    = (const float*)d_in[6];
    const float* b2    = (const float*)d_in[7];

    const int Nn   = in_sizes[3];            // 20000
    const int E    = in_sizes[2];            // 320000
    const int Fin  = in_sizes[0] / Nn;       // 512
    const int Fout = in_sizes[5];            // 1024
    const int G    = out_size / Fout;        // 64

    const int* src = ei;
    const int* dst = ei + E;

    auto alignup = [](size_t v) { return (v + 255) & ~(size_t)255; };
    char* ws = (char*)d_ws;
    const size_t bufBytes = alignup((size_t)Nn * Fout * sizeof(float));
    float* xw   = (float*)(ws);                       // [Nn,Fout] GEMM output
    float* h    = (float*)(ws + bufBytes);            // [Nn,Fout] aggregated/activated
    float* dinv = (float*)(ws + 2 * bufBytes);        // [Nn] deg -> dinv in place
    float* pool = (float*)(ws + 2 * bufBytes + alignup((size_t)Nn * sizeof(float)));
    float* cnt  = pool + (size_t)G * Fout;            // [G]
    (void)ws_size; (void)n_in;

    const int t256 = 256;
    const int fvec = Fout / 4;                        // 256 threads per feature row

    // --- symmetric normalization: deg = 1 (self loop) + sum_{e: dst} ew[e]; dinv = rsqrt(deg)
    gcn_fill<<<(Nn + t256 - 1) / t256, t256, 0, stream>>>(dinv, Nn, 1.0f);
    gcn_edge_deg<<<(E + t256 - 1) / t256, t256, 0, stream>>>(dst, ew, dinv, E);
    gcn_rsqrt_inplace<<<(Nn + t256 - 1) / t256, t256, 0, stream>>>(dinv, Nn);

    dim3 gemmBlock(256);
    dim3 gemmGrid1(Fout / TN, (Nn + TM - 1) / TM);

    // ----- layer 1: xw = x @ W1; h = norm-aggregate(xw) + b1, ReLU
    gcn_gemm_wmma_f32<<<gemmGrid1, gemmBlock, 0, stream>>>(x, W1, xw, Nn, Fin, Fout);
    gcn_self_init<<<Nn, fvec, 0, stream>>>(xw, dinv, h, Fout);
    gcn_edge_scatter<<<E, fvec, 0, stream>>>(src, dst, ew, dinv, xw, h, Fout);
    gcn_bias_relu<<<Nn, fvec, 0, stream>>>(h, b1, Fout);

    // ----- layer 2: xw = h @ W2; h = norm-aggregate(xw) + b2, ReLU  (ping-pong buffers)
    gcn_gemm_wmma_f32<<<gemmGrid1, gemmBlock, 0, stream>>>(h, W2, xw, Nn, Fout, Fout);
    gcn_self_init<<<Nn, fvec, 0, stream>>>(xw, dinv, h, Fout);
    gcn_edge_scatter<<<E, fvec, 0, stream>>>(src, dst, ew, dinv, xw, h, Fout);
    gcn_bias_relu<<<Nn, fvec, 0, stream>>>(h, b2, Fout);

    // ----- mean pool per graph
    int poolElems = G * Fout + G;                     // pool + cnt contiguous
    gcn_fill<<<(poolElems + t256 - 1) / t256, t256, 0, stream>>>(pool, poolElems, 0.0f);
    gcn_pool_accum<<<Nn, fvec, 0, stream>>>(h, batch, pool, cnt, Fout);
    gcn_pool_div<<<G, fvec, 0, stream>>>(pool, cnt, (float*)d_out, Fout);
}